// AttentionConv_42133629174422
// MI455X (gfx1250) — compile-verified
//
#include <hip/hip_runtime.h>
#include <hip/hip_bf16.h>
#include <math.h>

typedef float v2f __attribute__((ext_vector_type(2)));
typedef float v8f __attribute__((ext_vector_type(8)));

#define B_    8
#define CIN   512
#define COUT  512
#define L_    2048
#define KW    7
#define PAD_  3
#define NPAD  (L_ + 2 * PAD_)   // 2054 valid padded columns
#define LP    2064              // storage stride (multiple of 16, >= NPAD)
#define KC    64                // K-chunk staged in LDS
#define BPITCH 68               // LDS pitch: 8B-aligned b64 + conflict-free banks

// ---------------------------------------------------------------------------
// Stage 1: q/k/v = W{q,k,v} @ x over the padded column domain, via
// V_WMMA_F32_16X16X4_F32 (exact fp32 matrix math).
// Block = 256 threads (8 waves) computing a 128(M) x 64(N) tile.
// The 64x64 x-tile (B operand) is staged once per block into LDS,
// transposed + zero-masked, and shared by all 8 waves.
// qkv layout: [sel(3)][b][COUT][LP] in d_ws.
// ---------------------------------------------------------------------------
__global__ __launch_bounds__(256)
void qkv_gemm_f32wmma(const float* __restrict__ x,
                      const float* __restrict__ Wq,
                      const float* __restrict__ Wk,
                      const float* __restrict__ Wv,
                      float* __restrict__ qkv) {
    __shared__ float Bs[64 * BPITCH];    // [col][c] transposed, 17 KB

    const int sel = blockIdx.z % 3;
    const int b   = blockIdx.z / 3;
    const float* W = (sel == 0) ? Wq : (sel == 1) ? Wk : Wv;

    const int lane = threadIdx.x & 31;
    const int wv   = threadIdx.x >> 5;      // wave id 0..7
    const int lm   = lane & 15;
    const int half = lane >> 4;             // 0 or 1 (selects K pair)

    const int M0 = blockIdx.y * 128 + wv * 16;   // output row tile base
    const int n0 = blockIdx.x * 64;              // output col tile base (padded coords)

    // A fragment source: W row (M0 + lm), contiguous K pair -> b64 load
    const float* Wrow = W + (size_t)(M0 + lm) * CIN;

    // Staging mapping: thread -> (col = tid%64, rows tid/64 + 4*it)
    const int scol = threadIdx.x & 63;
    const int srow = threadIdx.x >> 6;           // 0..3
    int xc   = n0 + scol - PAD_;                 // padded col -> x col
    bool inb = (xc >= 0) && (xc < L_);
    const float* xsrc = x + (size_t)b * CIN * L_ + (inb ? xc : 0); // clamped
    const float smask = inb ? 1.0f : 0.0f;       // zero-pad applied ONCE at staging

    v8f acc[4];
#pragma unroll
    for (int t = 0; t < 4; ++t)
#pragma unroll
        for (int r = 0; r < 8; ++r) acc[t][r] = 0.0f;

    for (int c0 = 0; c0 < CIN; c0 += KC) {
        // ---- stage 64(K) x 64(N) tile, transposed, masked ----
#pragma unroll
        for (int it = 0; it < 16; ++it) {
            int cl = it * 4 + srow;
            Bs[scol * BPITCH + cl] = xsrc[(size_t)(c0 + cl) * L_] * smask;
        }
        __syncthreads();

        // ---- 64 WMMAs per wave on the staged tile ----
#pragma unroll 4
        for (int cl = 0; cl < KC; cl += 4) {
            const int cg = c0 + cl + 2 * half;
            v2f a = *(const v2f*)(Wrow + cg);            // global_load_b64
#pragma unroll
            for (int t = 0; t < 4; ++t) {
                // ds_load_b64: (K=c, K=c+1) contiguous at column (t*16+lm)
                v2f bb = *(const v2f*)(&Bs[(t * 16 + lm) * BPITCH + cl + 2 * half]);
                // (neg_a, A, neg_b, B, c_mod, C, reuse_a, reuse_b)
                acc[t] = __builtin_amdgcn_wmma_f32_16x16x4_f32(
                    false, a, false, bb, (short)0, acc[t], false, false);
            }
        }
        __syncthreads();
    }

    // C/D layout: n = lane%16, m = vgpr + 8*(lane/16)
    float* outp = qkv + (size_t)(sel * B_ + b) * COUT * LP;
#pragma unroll
    for (int t = 0; t < 4; ++t) {
        int col = n0 + t * 16 + lm;
        if (col < LP) {
#pragma unroll
            for (int r = 0; r < 8; ++r) {
                int m = M0 + r + 8 * half;
                outp[(size_t)m * LP + col] = acc[t][r];
            }
        }
    }
}

// ---------------------------------------------------------------------------
// Stage 2: 7-tap windowed softmax attention.
//   out[b,o,l] = sum_j softmax_j( q[l] * (k_pad[l+j] + rel[o,j]) ) * v_pad[l+j]
// q lives at padded column l+3. One thread per 4 consecutive l.
// ---------------------------------------------------------------------------
__global__ __launch_bounds__(256)
void win_attn(const float* __restrict__ qkv,
              const float* __restrict__ rel_h,
              const float* __restrict__ rel_w,
              float* __restrict__ out) {
    const int L4 = L_ / 4;
    int t = blockIdx.x * blockDim.x + threadIdx.x;
    if (t >= B_ * COUT * L4) return;

    int l4 = (t % L4) * 4;
    int o  = (t / L4) % COUT;
    int b  = t / (L4 * COUT);

    const float* qb = qkv + ((size_t)(0 * B_ + b) * COUT + o) * LP;
    const float* kb = qkv + ((size_t)(1 * B_ + b) * COUT + o) * LP;
    const float* vb = qkv + ((size_t)(2 * B_ + b) * COUT + o) * LP;

    float rel[KW];
    const float* rp = (o < COUT / 2) ? (rel_h + (size_t)o * KW)
                                     : (rel_w + (size_t)(o - COUT / 2) * KW);
#pragma unroll
    for (int j = 0; j < KW; ++j) rel[j] = rp[j];

    // 12-wide sliding windows, aligned float4 loads (l4 % 4 == 0, LP % 4 == 0)
    float kwin[12], vwin[12];
#pragma unroll
    for (int i = 0; i < 3; ++i) {
        float4 kk = *(const float4*)(kb + l4 + i * 4);
        float4 vv = *(const float4*)(vb + l4 + i * 4);
        kwin[i * 4 + 0] = kk.x; kwin[i * 4 + 1] = kk.y;
        kwin[i * 4 + 2] = kk.z; kwin[i * 4 + 3] = kk.w;
        vwin[i * 4 + 0] = vv.x; vwin[i * 4 + 1] = vv.y;
        vwin[i * 4 + 2] = vv.z; vwin[i * 4 + 3] = vv.w;
    }

    float4 res;
    float* resp = &res.x;
#pragma unroll
    for (int i = 0; i < 4; ++i) {
        float q = qb[l4 + i + PAD_];
        float lg[KW];
        float mx = -INFINITY;
#pragma unroll
        for (int j = 0; j < KW; ++j) {
            lg[j] = q * (kwin[i + j] + rel[j]);
            mx = fmaxf(mx, lg[j]);
        }
        float s = 0.0f, accv = 0.0f;
#pragma unroll
        for (int j = 0; j < KW; ++j) {
            float e = __expf(lg[j] - mx);
            s += e;
            accv += e * vwin[i + j];
        }
        resp[i] = accv / s;
    }
    *(float4*)(out + ((size_t)b * COUT + o) * L_ + l4) = res;
}

// ---------------------------------------------------------------------------
extern "C" void kernel_launch(void* const* d_in, const int* in_sizes, int n_in,
                              void* d_out, int out_size, void* d_ws, size_t ws_size,
                              hipStream_t stream) {
    const float* x     = (const float*)d_in[0];
    const float* Wq    = (const float*)d_in[1];
    const float* Wk    = (const float*)d_in[2];
    const float* Wv    = (const float*)d_in[3];
    const float* rel_h = (const float*)d_in[4];
    const float* rel_w = (const float*)d_in[5];
    float* qkv = (float*)d_ws;   // 3 * B * COUT * LP floats (~97 MiB)
    float* out = (float*)d_out;

    // 33 N-blocks of 64 cover padded cols 0..2111 (stores guarded to < LP)
    dim3 g1(33, COUT / 128, B_ * 3);
    qkv_gemm_f32wmma<<<g1, 256, 0, stream>>>(x, Wq, Wk, Wv, qkv);

    int total = B_ * COUT * (L_ / 4);
    win_attn<<<(total + 255) / 256, 256, 0, stream>>>(qkv, rel_h, rel_w, out);
}